// operators_52261162057769
// MI455X (gfx1250) — compile-verified
//
#include <hip/hip_runtime.h>
#include <math.h>

typedef float v2f __attribute__((ext_vector_type(2)));
typedef float v8f __attribute__((ext_vector_type(8)));

#define NN    128
#define PADS  136                 // LDS row stride: 2*136 % 64 == 16 -> conflict-free WMMA operand fetch
#define PLANE 16777216            // 16*64*128*128 elements per output plane

__device__ __forceinline__ v8f wmma4(v2f a, v2f b, v8f c) {
  // D = A(16x4, f32) * B(4x16, f32) + C(16x16, f32)
  return __builtin_amdgcn_wmma_f32_16x16x4_f32(false, a, false, b, (short)0, c, false, false);
}

// ---------------------------------------------------------------------------
// Setup: build the six 128x128 spectral-operator matrices into d_ws.
//   A  = IFFT diag(i*k)  FFT   (complex)  -> stored transposed (A-operand): LAr, LAi
//   A2 = IFFT diag(-k^2) FFT   (real)     -> LA2 (transposed)
//   B  = IFFT diag(-k)   FFT   (complex)  -> stored as B^T (B-operand): RBr, RBi
//   B2 = IFFT diag(+k^2) FFT   (real)     -> RB2 (as B2^T)
// Entry(out_idx=col, k_idx=row) depends only on d = col - row:
//   (1/N) * sum_t f(k_t) * e^{2*pi*i * t * d / N},  k_t = fftfreq(t)*N
// ---------------------------------------------------------------------------
__global__ void build_spectral_mats(float* __restrict__ mats) {
  int e = blockIdx.x * blockDim.x + threadIdx.x;
  if (e >= 6 * NN * NN) return;
  int mat = e >> 14;
  int ent = e & (NN * NN - 1);
  int row = ent >> 7;          // k index (matmul K dimension)
  int col = ent & 127;         // output (m or n) index
  int d = col - row;
  const float w0 = 6.28318530717958647692f / 128.0f;
  float acc = 0.0f;
  for (int t = 0; t < NN; ++t) {
    int kv = (t < 64) ? t : t - 128;            // jnp.fft.fftfreq(128)*128
    int r = (t * d) & 127;                       // exact angle reduction mod N
    float th = (float)r * w0;
    float sn = sinf(th), cs = cosf(th);
    float kvf = (float)kv;
    float k2f = (float)(kv * kv);
    float term;
    switch (mat) {
      case 0: term = -kvf * sn; break;           // Re A
      case 1: term =  kvf * cs; break;           // Im A
      case 2: term = -k2f * cs; break;           // A2 (real)
      case 3: term = -kvf * cs; break;           // Re B
      case 4: term = -kvf * sn; break;           // Im B
      default: term =  k2f * cs; break;          // B2 (real)
    }
    acc += term;
  }
  mats[e] = acc * (1.0f / 128.0f);
}

// ---------------------------------------------------------------------------
// Main kernel: one workgroup per (b,c) matrix. 256 threads = 8 waves.
// ---------------------------------------------------------------------------
__global__ void __launch_bounds__(256)
dil_frame_kernel(const float* __restrict__ x,
                 const float* __restrict__ df,
                 const float* __restrict__ d2f,
                 const float* __restrict__ mats,
                 float* __restrict__ out) {
  __shared__ float Xr[NN * PADS];   // X row-major (B-operand for left-mults)
  __shared__ float Xt[NN * PADS];   // X transposed (A-operand for right-mults)
  __shared__ float Zr[NN * PADS];   // Re(X * B^T)
  __shared__ float Zi[NN * PADS];   // Im(X * B^T)

  const int bc  = blockIdx.x;               // 0..1023 = b*64 + c
  const int tid = threadIdx.x;
  const float* __restrict__ Xg = x + (size_t)bc * (NN * NN);

  // Phase 1: stage X (and X^T) into LDS.
  for (int e = tid; e < NN * NN; e += 256) {
    int r = e >> 7, c = e & 127;
    float v = Xg[e];
    Xr[r * PADS + c] = v;
    Xt[c * PADS + r] = v;
  }
  __syncthreads();

  const int wave = tid >> 5;
  const int lane = tid & 31;
  const int ln   = lane & 15;
  const int g0   = (lane < 16) ? 0 : 2;   // K sub-index per lane half (16x4 operand layout)
  const int hi   = (lane < 16) ? 0 : 8;   // C/D row offset per lane half

  const float* __restrict__ LAr = mats;
  const float* __restrict__ LAi = mats + 16384;
  const float* __restrict__ LA2 = mats + 32768;
  const float* __restrict__ RBr = mats + 49152;
  const float* __restrict__ RBi = mats + 65536;
  const float* __restrict__ RB2 = mats + 81920;

  // Phase 2: Z = X * B^T (complex), 8 tiles of 16x16 per wave.
  for (int s = 0; s < 8; ++s) {
    int tt = wave * 8 + s;
    int tm = (tt >> 3) << 4;
    int tn = (tt & 7) << 4;
    v8f zr = {}; v8f zi = {};
    for (int k0 = 0; k0 < NN; k0 += 4) {
      int kk = k0 + g0;
      v2f a;  a.x  = Xt[kk * PADS + tm + ln];   a.y  = Xt[(kk + 1) * PADS + tm + ln];
      v2f br; br.x = RBr[kk * 128 + tn + ln];   br.y = RBr[(kk + 1) * 128 + tn + ln];
      v2f bi; bi.x = RBi[kk * 128 + tn + ln];   bi.y = RBi[(kk + 1) * 128 + tn + ln];
      zr = wmma4(a, br, zr);
      zi = wmma4(a, bi, zi);
    }
    for (int j = 0; j < 8; ++j) {
      int row = tm + j + hi;
      Zr[row * PADS + tn + ln] = zr[j];
      Zi[row * PADS + tn + ln] = zi[j];
    }
  }
  __syncthreads();

  // Per-channel complex parameters.
  const int ch = bc & 63;
  float dfre = df[2 * ch], dfim = df[2 * ch + 1];
  float ea   = expf(dfre);
  float dfcr = ea * cosf(dfim), dfci = ea * sinf(dfim);
  float d2fr = d2f[2 * ch], d2fi = d2f[2 * ch + 1];

  float* __restrict__ o0 = out + (size_t)bc * (NN * NN);

  // Phase 3: Y = A*X, P = A2*X, Q = X*B2^T, cross = A*Z; epilogue + stores.
  for (int s = 0; s < 8; ++s) {
    int tt = wave * 8 + s;
    int tm = (tt >> 3) << 4;
    int tn = (tt & 7) << 4;
    v8f yr = {}, yi = {}, p = {}, q = {};
    v8f crr = {}, cri = {}, cir = {}, cii = {};
    for (int k0 = 0; k0 < NN; k0 += 4) {
      int kk = k0 + g0;
      v2f bx;  bx.x  = Xr[kk * PADS + tn + ln];  bx.y  = Xr[(kk + 1) * PADS + tn + ln];
      v2f ar;  ar.x  = LAr[kk * 128 + tm + ln];  ar.y  = LAr[(kk + 1) * 128 + tm + ln];
      v2f ai;  ai.x  = LAi[kk * 128 + tm + ln];  ai.y  = LAi[(kk + 1) * 128 + tm + ln];
      v2f a2;  a2.x  = LA2[kk * 128 + tm + ln];  a2.y  = LA2[(kk + 1) * 128 + tm + ln];
      v2f ax;  ax.x  = Xt[kk * PADS + tm + ln];  ax.y  = Xt[(kk + 1) * PADS + tm + ln];
      v2f b2;  b2.x  = RB2[kk * 128 + tn + ln];  b2.y  = RB2[(kk + 1) * 128 + tn + ln];
      v2f bzr; bzr.x = Zr[kk * PADS + tn + ln];  bzr.y = Zr[(kk + 1) * PADS + tn + ln];
      v2f bzi; bzi.x = Zi[kk * PADS + tn + ln];  bzi.y = Zi[(kk + 1) * PADS + tn + ln];
      yr  = wmma4(ar, bx,  yr);
      yi  = wmma4(ai, bx,  yi);
      p   = wmma4(a2, bx,  p);
      q   = wmma4(ax, b2,  q);
      crr = wmma4(ar, bzr, crr);
      cri = wmma4(ar, bzi, cri);
      cir = wmma4(ai, bzr, cir);
      cii = wmma4(ai, bzi, cii);
    }
    for (int j = 0; j < 8; ++j) {
      int row = tm + j + hi;
      int col = tn + ln;
      int e   = row * 128 + col;
      float dxr = yr[j] + Zr[row * PADS + col];
      float dxi = yi[j] + Zi[row * PADS + col];
      float d2r = p[j] + q[j] + 2.0f * (crr[j] - cii[j]);
      float d2i = 2.0f * (cri[j] + cir[j]);

      float h = dxr * dxr + dxi * dxi;
      bool valid = h >= 1e-6f;                     // |dx0| >= 1e-3
      float sr = valid ? dxr : 1.0f;
      float si = valid ? dxi : 0.0f;
      float inv = 1.0f / (sr * sr + si * si);
      float a2r = (dfcr * sr + dfci * si) * inv;   // a2 = dfc / dx0s
      float a2i = (dfci * sr - dfcr * si) * inv;

      float aqr = a2r * a2r - a2i * a2i;           // a2^2
      float aqi = 2.0f * a2r * a2i;
      float invq = 1.0f / (aqr * aqr + aqi * aqi);
      float t1r = (d2fr * aqr + d2fi * aqi) * invq; // d2fc / a2^2
      float t1i = (d2fi * aqr - d2fr * aqi) * invq;
      float tr = d2r - t1r, ti = d2i - t1i;

      float s2r = sr * sr - si * si;               // dx0s^2
      float s2i = 2.0f * sr * si;
      float invs = 1.0f / (s2r * s2r + s2i * s2i);
      float wr = (dfcr * s2r + dfci * s2i) * invs; // dfc / dx0s^2
      float wi = (dfci * s2r - dfcr * s2i) * invs;
      float anr = 0.5f * (tr * wr - ti * wi);
      float ani = 0.5f * (tr * wi + ti * wr);

      float lnA = valid ? fminf(fmaxf(0.5f * logf(a2r * a2r + a2i * a2i), -3.0f), 3.0f) : 0.0f;
      float Phi = valid ? atan2f(a2i, a2r) : 0.0f;
      float am2 = anr * anr + ani * ani;
      bool tmask = valid && (am2 >= 1e-6f);
      float lnT = tmask ? fminf(fmaxf(0.5f * logf(am2), -2.59026717f), 3.0f) : -6.90775528f;
      float Psi = tmask ? atan2f(ani, anr) : 0.0f;

      o0[e]                       = h;
      o0[(size_t)PLANE + e]       = lnA;
      o0[2 * (size_t)PLANE + e]   = Phi;
      o0[3 * (size_t)PLANE + e]   = lnT;
      o0[4 * (size_t)PLANE + e]   = Psi;
    }
  }
}

extern "C" void kernel_launch(void* const* d_in, const int* in_sizes, int n_in,
                              void* d_out, int out_size, void* d_ws, size_t ws_size,
                              hipStream_t stream) {
  const float* x   = (const float*)d_in[0];   // (16,64,128,128) f32
  const float* df  = (const float*)d_in[1];   // (1,64,2)
  const float* d2f = (const float*)d_in[2];   // (1,64,2)
  float* out  = (float*)d_out;                // 5 planes of 16,777,216 f32
  float* mats = (float*)d_ws;                 // 6 * 16384 f32 = 384 KB scratch
  if (ws_size < (size_t)6 * 16384 * sizeof(float)) return;

  build_spectral_mats<<<(6 * 16384 + 255) / 256, 256, 0, stream>>>(mats);
  dil_frame_kernel<<<1024, 256, 0, stream>>>(x, df, d2f, mats, out);
}